// mix_3D2D_encoding_721554505777
// MI455X (gfx1250) — compile-verified
//
#include <hip/hip_runtime.h>
#include <stdint.h>

// MI455X / gfx1250 hash-grid encoder.
// Strategy:
//   1) pack_kernel: binarize tables (sign in {-1,0,+1}) into 2 bits/feature ->
//      1 byte/entry. Packed tables = 7.9 MB total (fully L2-resident).
//   2) encode_kernel: 1 point/thread. TDM (tensor_load_to_lds) stages the three
//      smallest dense 3D levels (54 KB packed) into LDS; gathers are 1-byte
//      ds_load_u8 / global_load_u8; output written with non-temporal float4
//      stores so the 192 MB output stream does not evict packed tables from L2.

typedef float v4f __attribute__((ext_vector_type(4)));
typedef unsigned int u32x4 __attribute__((ext_vector_type(4)));
typedef int i32x8 __attribute__((ext_vector_type(8)));
typedef int i32x4 __attribute__((ext_vector_type(4)));

#define T3N 524288u   // 2^19 entries per 3D level
#define T2N 131072u   // 2^17 entries per 2D level
#define P1 2654435761u
#define P2 805459861u

__device__ constexpr unsigned RES3[12] = {18u,24u,33u,44u,59u,80u,108u,148u,201u,275u,376u,514u};
__device__ constexpr unsigned RES2[4]  = {130u,258u,514u,1026u};
// Staged 3D levels 0..2 (dense, 18^3/24^3/33^3 bytes packed)
__device__ constexpr unsigned SLDS[3] = {0u, 5832u, 19656u};     // LDS byte offsets
__device__ constexpr unsigned SQW[3]  = {729u, 1728u, 4493u};    // 8-byte units (33^3=35937 -> 35944)
#define STAGE_BYTES 55616                                        // 19656+35944=55600, padded

#if defined(__has_builtin)
#if __has_builtin(__builtin_amdgcn_tensor_load_to_lds) && __has_builtin(__builtin_amdgcn_s_wait_tensorcnt)
#define USE_TDM 1
#endif
#endif
#ifndef USE_TDM
#define USE_TDM 0
#endif

__device__ __forceinline__ unsigned umin_(unsigned a, unsigned b) { return a < b ? a : b; }

// decode 2-bit code -> sign value {-1,0,+1} as float
__device__ __forceinline__ float sgnf(unsigned byte, int f) {
  unsigned c = (byte >> (2 * f)) & 3u;
  return (float)((int)(c & 1u) - (int)((c >> 1) & 1u));
}
__device__ __forceinline__ float fsign(float v) {
  return (float)((int)(v > 0.f) - (int)(v < 0.f));
}
__device__ __forceinline__ unsigned sign_code(float f) {
  return (f > 0.f) ? 1u : ((f < 0.f) ? 2u : 0u);
}

// ---------------------------------------------------------------------------
// Tensor Data Mover: 1D copy of n_qw 8-byte elements from global to LDS.
// Descriptor built per CDNA5 ISA D# layout (CDNA5 ISA).
// This toolchain's builtin is the 6-argument form:
//   (v4u g0, v8i g1, v4i g2, v4i g3, v8i extra, i32 cpol)
// ---------------------------------------------------------------------------
__device__ __forceinline__ void tdm_load_1d(const void* gsrc, unsigned lds_addr, unsigned n_qw) {
#if USE_TDM
  unsigned long long ga = (unsigned long long)(uintptr_t)gsrc;
  u32x4 g0;
  g0[0] = 1u;                                   // count=1 valid, user mode, no gather
  g0[1] = lds_addr;                             // lds_addr (bytes)
  g0[2] = (unsigned)(ga & 0xffffffffull);       // global_addr[31:0]
  g0[3] = (unsigned)((ga >> 32) & 0x1ffffffull) // global_addr[56:32]
        | (2u << 30);                           // type = 2 ("image")
  i32x8 g1;
  g1[0] = (int)(3u << 16);                      // wg_mask=0, data_size=3 (8B), no flags
  g1[1] = (int)((n_qw & 0xffffu) << 16);        // atomic_addr=0 | tensor_dim0[15:0]
  g1[2] = (int)(((n_qw >> 16) & 0xffffu) | (1u << 16)); // tensor_dim0[31:16] | tensor_dim1.lo=1
  g1[3] = (int)((n_qw & 0xffffu) << 16);        // tensor_dim1.hi=0 | tile_dim0=n_qw
  g1[4] = 1;                                    // tile_dim1=1 | tile_dim2=0
  g1[5] = (int)n_qw;                            // tensor_dim0_stride[31:0]
  g1[6] = 0;                                    // stride hi | tensor_dim1_stride lo
  g1[7] = 0;
  i32x4 gz4 = {0, 0, 0, 0};                     // groups 2/3 unused (<=2D tensor)
  i32x8 gz8 = {0, 0, 0, 0, 0, 0, 0, 0};
  __builtin_amdgcn_tensor_load_to_lds(g0, g1, gz4, gz4, gz8, 0);
#else
  (void)gsrc; (void)lds_addr; (void)n_qw;
#endif
}

// ---------------------------------------------------------------------------
// Pack kernel: table entry (4 f32) -> 1 byte (2-bit sign codes)
// ---------------------------------------------------------------------------
__global__ __launch_bounds__(256) void pack_kernel(const v4f* __restrict__ src,
                                                   uint8_t* __restrict__ dst, int n) {
  int i = (int)(blockIdx.x * blockDim.x + threadIdx.x);
  if (i >= n) return;
  v4f v = src[i];
  unsigned b = sign_code(v.x) | (sign_code(v.y) << 2) | (sign_code(v.z) << 4) | (sign_code(v.w) << 6);
  dst[i] = (uint8_t)b;
}

// ---------------------------------------------------------------------------
// 2D level encoder over packed tables (4 levels, 4 corners each)
// ---------------------------------------------------------------------------
__device__ __forceinline__ void enc2d(float pa, float pb, const uint8_t* __restrict__ pk,
                                      v4f* __restrict__ outv) {
#pragma unroll
  for (int l = 0; l < 4; ++l) {
    const unsigned R = RES2[l];
    const unsigned Rm1 = R - 1u;
    float posa = pa * (float)Rm1; float a0f = floorf(posa); float fa = posa - a0f;
    float posb = pb * (float)Rm1; float b0f = floorf(posb); float fb = posb - b0f;
    unsigned ia = (unsigned)a0f, ib = (unsigned)b0f;
    unsigned a0 = umin_(ia, Rm1), a1 = umin_(ia + 1u, Rm1);
    unsigned b0 = umin_(ib, Rm1), b1 = umin_(ib + 1u, Rm1);
    const bool dense = (R * R <= T2N);
    const uint8_t* base = pk + (size_t)l * T2N;
    float s0 = 0.f, s1 = 0.f, s2 = 0.f, s3 = 0.f;
#pragma unroll
    for (int cb = 0; cb < 2; ++cb)
#pragma unroll
      for (int ca = 0; ca < 2; ++ca) {
        unsigned ga = ca ? a1 : a0;
        unsigned gb = cb ? b1 : b0;
        unsigned id = dense ? (ga + gb * R) : ((ga ^ (gb * P1)) & (T2N - 1u));
        unsigned byte = base[id];
        float w = (ca ? fa : 1.f - fa) * (cb ? fb : 1.f - fb);
        s0 = fmaf(sgnf(byte, 0), w, s0);
        s1 = fmaf(sgnf(byte, 1), w, s1);
        s2 = fmaf(sgnf(byte, 2), w, s2);
        s3 = fmaf(sgnf(byte, 3), w, s3);
      }
    v4f o = {s0, s1, s2, s3};
    __builtin_nontemporal_store(o, outv + l);
  }
}

// ---------------------------------------------------------------------------
// Main encoder (packed path). 1 point per thread, 256 threads/block.
// ---------------------------------------------------------------------------
__global__ __launch_bounds__(256) void encode_kernel(
    const float* __restrict__ x,
    const uint8_t* __restrict__ pk_xyz, const uint8_t* __restrict__ pk_xy,
    const uint8_t* __restrict__ pk_xz,  const uint8_t* __restrict__ pk_yz,
    float* __restrict__ out, int N) {
  __shared__ alignas(16) uint8_t s_tab[STAGE_BYTES];

#if USE_TDM
  if (threadIdx.x < 32u) {  // wave 0 issues the TDM DMAs (EXEC-independent)
    unsigned lds0 = (unsigned)(uintptr_t)(&s_tab[0]);
    tdm_load_1d(pk_xyz + 0ull * T3N, lds0 + SLDS[0], SQW[0]);
    tdm_load_1d(pk_xyz + 1ull * T3N, lds0 + SLDS[1], SQW[1]);
    tdm_load_1d(pk_xyz + 2ull * T3N, lds0 + SLDS[2], SQW[2]);
    __builtin_amdgcn_s_wait_tensorcnt(0);
  }
#else
  {
    const unsigned nb0 = 5832u, nb1 = 13824u, nb2 = 35937u;
    for (unsigned i = threadIdx.x; i < nb0; i += blockDim.x) s_tab[SLDS[0] + i] = pk_xyz[0u * T3N + i];
    for (unsigned i = threadIdx.x; i < nb1; i += blockDim.x) s_tab[SLDS[1] + i] = pk_xyz[1u * T3N + i];
    for (unsigned i = threadIdx.x; i < nb2; i += blockDim.x) s_tab[SLDS[2] + i] = pk_xyz[2u * T3N + i];
  }
#endif
  __syncthreads();

  int n = (int)(blockIdx.x * blockDim.x + threadIdx.x);
  if (n >= N) return;

  float px = __builtin_nontemporal_load(x + 3ull * (size_t)n + 0);
  float py = __builtin_nontemporal_load(x + 3ull * (size_t)n + 1);
  float pz = __builtin_nontemporal_load(x + 3ull * (size_t)n + 2);
  v4f* outv = (v4f*)(out + (size_t)n * 96u);

  // ---- 12 3D levels (8 corners each) ----
#pragma unroll
  for (int l = 0; l < 12; ++l) {
    const unsigned R = RES3[l];
    const unsigned Rm1 = R - 1u;
    float posx = px * (float)Rm1; float fx0 = floorf(posx); float fx = posx - fx0;
    float posy = py * (float)Rm1; float fy0 = floorf(posy); float fy = posy - fy0;
    float posz = pz * (float)Rm1; float fz0 = floorf(posz); float fz = posz - fz0;
    unsigned ix = (unsigned)fx0, iy = (unsigned)fy0, iz = (unsigned)fz0;
    unsigned x0 = umin_(ix, Rm1), x1 = umin_(ix + 1u, Rm1);
    unsigned y0 = umin_(iy, Rm1), y1 = umin_(iy + 1u, Rm1);
    unsigned z0 = umin_(iz, Rm1), z1 = umin_(iz + 1u, Rm1);
    const bool dense = ((unsigned long long)R * R * R <= (unsigned long long)T3N);
    const bool staged = (l < 3);
    const uint8_t* gbase = pk_xyz + (size_t)l * T3N;
    const unsigned lds_off = staged ? SLDS[l] : 0u;
    const unsigned RR = R * R;
    float s0 = 0.f, s1 = 0.f, s2 = 0.f, s3 = 0.f;
#pragma unroll
    for (int cz = 0; cz < 2; ++cz)
#pragma unroll
      for (int cy = 0; cy < 2; ++cy)
#pragma unroll
        for (int cx = 0; cx < 2; ++cx) {
          unsigned gx = cx ? x1 : x0;
          unsigned gy = cy ? y1 : y0;
          unsigned gz = cz ? z1 : z0;
          unsigned id = dense ? (gx + gy * R + gz * RR)
                              : ((gx ^ (gy * P1) ^ (gz * P2)) & (T3N - 1u));
          unsigned byte = staged ? (unsigned)s_tab[lds_off + id] : (unsigned)gbase[id];
          float w = (cx ? fx : 1.f - fx) * (cy ? fy : 1.f - fy) * (cz ? fz : 1.f - fz);
          s0 = fmaf(sgnf(byte, 0), w, s0);
          s1 = fmaf(sgnf(byte, 1), w, s1);
          s2 = fmaf(sgnf(byte, 2), w, s2);
          s3 = fmaf(sgnf(byte, 3), w, s3);
        }
    v4f o = {s0, s1, s2, s3};
    __builtin_nontemporal_store(o, outv + l);
  }

  // ---- 2D levels: xy, xz, yz (columns 48.., 64.., 80..) ----
  enc2d(px, py, pk_xy, outv + 12);
  enc2d(px, pz, pk_xz, outv + 16);
  enc2d(py, pz, pk_yz, outv + 20);
}

// ---------------------------------------------------------------------------
// Fallback encoder reading raw f32 tables (used only if ws too small to pack)
// ---------------------------------------------------------------------------
__device__ __forceinline__ void enc2d_f32(float pa, float pb, const v4f* __restrict__ tb,
                                          v4f* __restrict__ outv) {
#pragma unroll
  for (int l = 0; l < 4; ++l) {
    const unsigned R = RES2[l];
    const unsigned Rm1 = R - 1u;
    float posa = pa * (float)Rm1; float a0f = floorf(posa); float fa = posa - a0f;
    float posb = pb * (float)Rm1; float b0f = floorf(posb); float fb = posb - b0f;
    unsigned ia = (unsigned)a0f, ib = (unsigned)b0f;
    unsigned a0 = umin_(ia, Rm1), a1 = umin_(ia + 1u, Rm1);
    unsigned b0 = umin_(ib, Rm1), b1 = umin_(ib + 1u, Rm1);
    const bool dense = (R * R <= T2N);
    const v4f* base = tb + (size_t)l * T2N;
    float s0 = 0.f, s1 = 0.f, s2 = 0.f, s3 = 0.f;
#pragma unroll
    for (int cb = 0; cb < 2; ++cb)
#pragma unroll
      for (int ca = 0; ca < 2; ++ca) {
        unsigned ga = ca ? a1 : a0;
        unsigned gb = cb ? b1 : b0;
        unsigned id = dense ? (ga + gb * R) : ((ga ^ (gb * P1)) & (T2N - 1u));
        v4f e = base[id];
        float w = (ca ? fa : 1.f - fa) * (cb ? fb : 1.f - fb);
        s0 = fmaf(fsign(e.x), w, s0);
        s1 = fmaf(fsign(e.y), w, s1);
        s2 = fmaf(fsign(e.z), w, s2);
        s3 = fmaf(fsign(e.w), w, s3);
      }
    v4f o = {s0, s1, s2, s3};
    __builtin_nontemporal_store(o, outv + l);
  }
}

__global__ __launch_bounds__(256) void encode_f32_kernel(
    const float* __restrict__ x,
    const v4f* __restrict__ t_xyz, const v4f* __restrict__ t_xy,
    const v4f* __restrict__ t_xz,  const v4f* __restrict__ t_yz,
    float* __restrict__ out, int N) {
  int n = (int)(blockIdx.x * blockDim.x + threadIdx.x);
  if (n >= N) return;
  float px = x[3ull * (size_t)n + 0];
  float py = x[3ull * (size_t)n + 1];
  float pz = x[3ull * (size_t)n + 2];
  v4f* outv = (v4f*)(out + (size_t)n * 96u);
#pragma unroll
  for (int l = 0; l < 12; ++l) {
    const unsigned R = RES3[l];
    const unsigned Rm1 = R - 1u;
    float posx = px * (float)Rm1; float fx0 = floorf(posx); float fx = posx - fx0;
    float posy = py * (float)Rm1; float fy0 = floorf(posy); float fy = posy - fy0;
    float posz = pz * (float)Rm1; float fz0 = floorf(posz); float fz = posz - fz0;
    unsigned ix = (unsigned)fx0, iy = (unsigned)fy0, iz = (unsigned)fz0;
    unsigned x0 = umin_(ix, Rm1), x1 = umin_(ix + 1u, Rm1);
    unsigned y0 = umin_(iy, Rm1), y1 = umin_(iy + 1u, Rm1);
    unsigned z0 = umin_(iz, Rm1), z1 = umin_(iz + 1u, Rm1);
    const bool dense = ((unsigned long long)R * R * R <= (unsigned long long)T3N);
    const v4f* base = t_xyz + (size_t)l * T3N;
    const unsigned RR = R * R;
    float s0 = 0.f, s1 = 0.f, s2 = 0.f, s3 = 0.f;
#pragma unroll
    for (int cz = 0; cz < 2; ++cz)
#pragma unroll
      for (int cy = 0; cy < 2; ++cy)
#pragma unroll
        for (int cx = 0; cx < 2; ++cx) {
          unsigned gx = cx ? x1 : x0;
          unsigned gy = cy ? y1 : y0;
          unsigned gz = cz ? z1 : z0;
          unsigned id = dense ? (gx + gy * R + gz * RR)
                              : ((gx ^ (gy * P1) ^ (gz * P2)) & (T3N - 1u));
          v4f e = base[id];
          float w = (cx ? fx : 1.f - fx) * (cy ? fy : 1.f - fy) * (cz ? fz : 1.f - fz);
          s0 = fmaf(fsign(e.x), w, s0);
          s1 = fmaf(fsign(e.y), w, s1);
          s2 = fmaf(fsign(e.z), w, s2);
          s3 = fmaf(fsign(e.w), w, s3);
        }
    v4f o = {s0, s1, s2, s3};
    __builtin_nontemporal_store(o, outv + l);
  }
  enc2d_f32(px, py, t_xy, outv + 12);
  enc2d_f32(px, pz, t_xz, outv + 16);
  enc2d_f32(py, pz, t_yz, outv + 20);
}

// ---------------------------------------------------------------------------
extern "C" void kernel_launch(void* const* d_in, const int* in_sizes, int n_in,
                              void* d_out, int out_size, void* d_ws, size_t ws_size,
                              hipStream_t stream) {
  const float* x    = (const float*)d_in[0];
  const float* txyz = (const float*)d_in[1];
  const float* txy  = (const float*)d_in[2];
  const float* txz  = (const float*)d_in[3];
  const float* tyz  = (const float*)d_in[4];
  float* out = (float*)d_out;
  const int N = in_sizes[0] / 3;

  const size_t SZ3 = 12ull * (size_t)T3N;            // 6,291,456 B packed xyz
  const size_t SZ2 = 4ull * (size_t)T2N;             //   524,288 B per packed 2D set
  const size_t WS_NEEDED = SZ3 + 3ull * SZ2;         // 7,864,320 B

  if (ws_size >= WS_NEEDED) {
    uint8_t* ws = (uint8_t*)d_ws;
    uint8_t* pk_xyz = ws;
    uint8_t* pk_xy  = ws + SZ3;
    uint8_t* pk_xz  = pk_xy + SZ2;
    uint8_t* pk_yz  = pk_xz + SZ2;
    const int n3 = (int)(12u * T3N);
    const int n2 = (int)(4u * T2N);
    pack_kernel<<<(n3 + 255) / 256, 256, 0, stream>>>((const v4f*)txyz, pk_xyz, n3);
    pack_kernel<<<(n2 + 255) / 256, 256, 0, stream>>>((const v4f*)txy,  pk_xy,  n2);
    pack_kernel<<<(n2 + 255) / 256, 256, 0, stream>>>((const v4f*)txz,  pk_xz,  n2);
    pack_kernel<<<(n2 + 255) / 256, 256, 0, stream>>>((const v4f*)tyz,  pk_yz,  n2);
    encode_kernel<<<(N + 255) / 256, 256, 0, stream>>>(x, pk_xyz, pk_xy, pk_xz, pk_yz, out, N);
  } else {
    encode_f32_kernel<<<(N + 255) / 256, 256, 0, stream>>>(
        x, (const v4f*)txyz, (const v4f*)txy, (const v4f*)txz, (const v4f*)tyz, out, N);
  }
}